// TemporalSelfAttensionLayer_84215718739980
// MI455X (gfx1250) — compile-verified
//
#include <hip/hip_runtime.h>
#include <hip/hip_bf16.h>
#include <math.h>

typedef __attribute__((ext_vector_type(2))) float v2f;
typedef __attribute__((ext_vector_type(8))) float v8f;

#define TT 4096
#define DD 256
#define KD 128
#define OD 396
#define NROWS (4 * TT)          // B*T = 16384
#define PADV (-10000000.0f)

// ---------------------------------------------------------------------------
// Projection GEMM: out[NROWS x N] = x[NROWS x 256] * W[256 x N] + bias.
// One 16x16 output tile per wave via V_WMMA_F32_16X16X4_F32 (fp32 pipe,
// exact fp32 accumulation). N is compile-time so all loop addresses fold
// into immediate offsets; out-of-range columns (V's 396 -> 25 tiles) are
// handled by clamping the load column: WMMA output column n only consumes
// B column n, so clamped-lane garbage lands solely in unstored columns.
//   grid = (NROWS/16, ceil(N/16)), block = 32 (one wave)
// ---------------------------------------------------------------------------
template<int N>
__global__ __launch_bounds__(32) void proj_wmma(
    const float* __restrict__ x, const float* __restrict__ W,
    const float* __restrict__ bias, float* __restrict__ outp)
{
    const int lane  = threadIdx.x;        // 0..31 (wave32)
    const int half  = lane >> 4;          // 0: K pair {0,1}; 1: K pair {2,3}
    const int l16   = lane & 15;
    const int mTile = blockIdx.x;
    const int nt    = blockIdx.y;

    const int m  = mTile * 16 + l16;      // A-matrix row for this lane
    const int n  = nt * 16 + l16;         // B/C column for this lane
    const int nc = (n < N) ? n : (N - 1); // clamp (no divergence in loop)

    const float* __restrict__ xp = x + (size_t)m * DD + 2 * half;
    const float* __restrict__ wp = W + (size_t)(2 * half) * N + nc;

    v8f c = {};
    #pragma unroll 8
    for (int kk = 0; kk < DD; kk += 4) {
        v2f a;                      // A: row m, K elems {ka, ka+1} (one b64)
        a.x = xp[0];
        a.y = xp[1];
        v2f b;                      // B: column n, same K striping
        b.x = wp[0];
        b.y = wp[N];
        c = __builtin_amdgcn_wmma_f32_16x16x4_f32(
                /*neg_a=*/false, a, /*neg_b=*/false, b,
                /*c_mod=*/(short)0, c, /*reuse_a=*/false, /*reuse_b=*/false);
        xp += 4;
        wp += 4 * N;
    }

    if (n < N) {
        const float bb = bias[n];
        // C/D layout: VGPR r -> (M = r + 8*half, N = l16)
        #pragma unroll
        for (int r = 0; r < 8; ++r) {
            const int mm = mTile * 16 + r + 8 * half;
            outp[(size_t)mm * N + n] = c[r] + bb;
        }
    }
}

// ---------------------------------------------------------------------------
// Banded (±3) attention. One query row per wave32.
//   grid.x = NROWS/8, block = 256 (8 waves)
// ---------------------------------------------------------------------------
__global__ __launch_bounds__(256) void band_attn(
    const float* __restrict__ q, const float* __restrict__ k,
    const float* __restrict__ v, float* __restrict__ out)
{
    const int lane = threadIdx.x & 31;
    const int wid  = threadIdx.x >> 5;
    const int row  = blockIdx.x * 8 + wid;   // 0..NROWS-1
    if (row >= NROWS) return;

    const int t    = row & (TT - 1);         // position within batch
    const int base = row - t;                // batch start row

    const float* __restrict__ qrow = q + (size_t)row * KD;
    const float q0 = qrow[lane];
    const float q1 = qrow[lane + 32];
    const float q2 = qrow[lane + 64];
    const float q3 = qrow[lane + 96];

    const float scale = 0.08838834764831845f;   // 1/sqrt(128)

    float s[7];
    #pragma unroll
    for (int d = 0; d < 7; ++d) {
        const int j = t + d - 3;
        const bool ok = (j >= 0) && (j < TT);
        float dot = 0.0f;
        if (ok) {
            const float* __restrict__ krow = k + (size_t)(base + j) * KD;
            dot = fmaf(q0, krow[lane],
                  fmaf(q1, krow[lane + 32],
                  fmaf(q2, krow[lane + 64], q3 * krow[lane + 96])));
        }
        // wave32 butterfly reduction
        #pragma unroll
        for (int off = 16; off > 0; off >>= 1)
            dot += __shfl_xor(dot, off, 32);
        const float sv = ok ? dot * scale : 0.0f;
        // faithful to reference: |sim|==0 entries (incl. out-of-band) get PAD
        s[d] = (fabsf(sv) > 0.0f) ? sv : PADV;
    }

    float m = s[0];
    #pragma unroll
    for (int d = 1; d < 7; ++d) m = fmaxf(m, s[d]);

    float p[7], sum = 0.0f;
    #pragma unroll
    for (int d = 0; d < 7; ++d) { p[d] = expf(s[d] - m); sum += p[d]; }
    const float inv = 1.0f / sum;
    #pragma unroll
    for (int d = 0; d < 7; ++d) p[d] *= inv;

    float* __restrict__ orow = out + (size_t)row * OD;
    for (int n = lane; n < OD; n += 32) {
        float acc = 0.0f;
        #pragma unroll
        for (int d = 0; d < 7; ++d) {
            const int j = t + d - 3;
            if (j >= 0 && j < TT)
                acc = fmaf(p[d], v[(size_t)(base + j) * OD + n], acc);
        }
        orow[n] = acc;
    }
}

// ---------------------------------------------------------------------------
extern "C" void kernel_launch(void* const* d_in, const int* in_sizes, int n_in,
                              void* d_out, int out_size, void* d_ws, size_t ws_size,
                              hipStream_t stream)
{
    (void)in_sizes; (void)n_in; (void)out_size; (void)ws_size;

    const float* x  = (const float*)d_in[0];
    const float* wq = (const float*)d_in[1];
    const float* bq = (const float*)d_in[2];
    const float* wk = (const float*)d_in[3];
    const float* bk = (const float*)d_in[4];
    const float* wv = (const float*)d_in[5];
    const float* bv = (const float*)d_in[6];
    float* out = (float*)d_out;

    float* qw = (float*)d_ws;                       // NROWS*128 floats (8 MiB)
    float* kw = qw + (size_t)NROWS * KD;            // NROWS*128 floats (8 MiB)
    float* vw = kw + (size_t)NROWS * KD;            // NROWS*396 floats (24.75 MiB)

    // Projections: one wave per 16x16 output tile
    proj_wmma<KD><<<dim3(NROWS / 16, KD / 16),       32, 0, stream>>>(x, wq, bq, qw);
    proj_wmma<KD><<<dim3(NROWS / 16, KD / 16),       32, 0, stream>>>(x, wk, bk, kw);
    proj_wmma<OD><<<dim3(NROWS / 16, (OD + 15) / 16), 32, 0, stream>>>(x, wv, bv, vw);

    // Banded attention: one wave32 per query row, 8 waves per block
    band_attn<<<dim3(NROWS / 8), 256, 0, stream>>>(qw, kw, vw, out);
}